// FCOSTarget_22256520528444
// MI455X (gfx1250) — compile-verified
//
#include <hip/hip_runtime.h>
#include <math.h>

// ---------------------------------------------------------------------------
// FCOS target assignment for gfx1250.
// One wave32 per 16-location tile. Offsets (l,t,r,b) for a 16-loc x 16-box
// tile are computed with V_WMMA_F32_16X16X4_F32 as an outer product:
//   l = x*1 + y*0 + 1*(-x1) ; t = y - y1 ; r = x2 - x ; b = y2 - y
// A[16x4] = per-location (x, y, 1, 0);  B[4x16] = per-box coefficients.
// ---------------------------------------------------------------------------

typedef float v2f __attribute__((ext_vector_type(2)));
typedef float v8f __attribute__((ext_vector_type(8)));

#define INF_AREA 99999999.0f
#define TILES_PER_BATCH 1067   // 800 + 200 + 50 + 13 + 4

// per-level constants (levels: strides 8,16,32,64,128)
__constant__ int   kTileStart[5] = {0, 800, 1000, 1050, 1063};
__constant__ int   kHW[5]        = {12800, 3200, 800, 208, 56};
__constant__ int   kW[5]         = {128, 64, 32, 16, 8};
__constant__ float kStride[5]    = {8.f, 16.f, 32.f, 64.f, 128.f};
__constant__ float kRmin[5]      = {-1.f, 64.f, 128.f, 256.f, 512.f};
__constant__ float kRmax[5]      = {64.f, 128.f, 256.f, 512.f, 100000000.f};
// flat-float offsets into d_out: cls[5] | reg[5] | ctr[5]
__constant__ int   kClsOff[5]    = {0, 102400, 128000, 134400, 136064};
__constant__ int   kRegOff[5]    = {136512, 546112, 648512, 674112, 680768};
__constant__ int   kCtrOff[5]    = {682560, 784960, 810560, 816960, 818624};

__global__ __launch_bounds__(32)
void fcos_target_wmma(const float* __restrict__ boxes,   // [8,64,4]
                      const int*   __restrict__ cls_ids, // [8,64]
                      float*       __restrict__ out)
{
    __shared__ float4 sBox[64];
    __shared__ int    sCls[64];

    const int lane = threadIdx.x;          // 0..31
    const int b    = blockIdx.x / TILES_PER_BATCH;
    const int tile = blockIdx.x - b * TILES_PER_BATCH;

    // level from tile id (uniform across wave)
    const int level = (tile >= 800) + (tile >= 1000) + (tile >= 1050) + (tile >= 1063);
    const int   tIn    = tile - kTileStart[level];
    const int   hw     = kHW[level];
    const int   w      = kW[level];
    const float stride = kStride[level];
    const float rmin   = kRmin[level];
    const float rmax   = kRmax[level];

    // ---- stage boxes + class ids for this batch into LDS ----
    const float4* gBox = (const float4*)(boxes + (size_t)b * 64 * 4);
    sBox[lane]      = gBox[lane];
    sBox[lane + 32] = gBox[lane + 32];
    sCls[lane]      = cls_ids[b * 64 + lane];
    sCls[lane + 32] = cls_ids[b * 64 + lane + 32];
    __syncthreads();

    const int half = lane >> 4;            // 0: rows 0-7 of C/D, 1: rows 8-15
    const int l16  = lane & 15;

    // ---- A matrix: 16x4 f32, lanes 0-15 supply K=0,1 (x,y); lanes 16-31 K=2,3 (1,0)
    const int   locA = tIn * 16 + l16;     // location for A-row l16 (may be padded)
    const float xs   = (float)(locA % w) * stride + stride * 0.5f;
    const float ys   = (float)(locA / w) * stride + stride * 0.5f;
    v2f A;
    A.x = half ? 1.0f : xs;
    A.y = half ? 0.0f : ys;

    float bA[8];
    int   bI[8];
#pragma unroll
    for (int v = 0; v < 8; ++v) { bA[v] = INF_AREA; bI[v] = 0; }

    const v8f   czero     = {};
    const float ctrThresh = 3.0f * stride; // 2 * SAMPLE_RADIO * stride

    // ---- 4 box tiles of 16: 4 WMMAs each (l,t,r,b), then mask/area/argmin ----
#pragma unroll
    for (int t = 0; t < 4; ++t) {
        const int    n  = t * 16 + l16;    // this lane's box column
        const float4 bx = sBox[n];

        // B matrix 4x16: lanes 0-15 supply K=0,1 rows; lanes 16-31 K=2,3 rows
        v2f Bl, Bt, Br, Bb;
        Bl.x = half ? -bx.x : 1.0f;  Bl.y = 0.0f;                 // l = x - x1
        Bt.x = half ? -bx.y : 0.0f;  Bt.y = half ? 0.0f : 1.0f;   // t = y - y1
        Br.x = half ?  bx.z : -1.0f; Br.y = 0.0f;                 // r = x2 - x
        Bb.x = half ?  bx.w : 0.0f;  Bb.y = half ? 0.0f : -1.0f;  // b = y2 - y

        v8f L  = __builtin_amdgcn_wmma_f32_16x16x4_f32(false, A, false, Bl, (short)0, czero, false, false);
        v8f T  = __builtin_amdgcn_wmma_f32_16x16x4_f32(false, A, false, Bt, (short)0, czero, false, false);
        v8f R  = __builtin_amdgcn_wmma_f32_16x16x4_f32(false, A, false, Br, (short)0, czero, false, false);
        v8f Bo = __builtin_amdgcn_wmma_f32_16x16x4_f32(false, A, false, Bb, (short)0, czero, false, false);

#pragma unroll
        for (int v = 0; v < 8; ++v) {
            // element (location m = v + 8*half, box n)
            float l  = L[v], tt = T[v], r = R[v], bb = Bo[v];
            float omin = fminf(fminf(l, tt), fminf(r, bb));
            float omax = fmaxf(fmaxf(l, tt), fmaxf(r, bb));
            float cd   = fmaxf(fabsf(l - r), fabsf(tt - bb)); // 2*max center offset
            bool  pos  = (omin > 0.0f) && (omax > rmin) && (omax <= rmax) && (cd <= ctrThresh);
            float area = pos ? (l + r) * (tt + bb) : INF_AREA;
            if (area < bA[v]) { bA[v] = area; bI[v] = n; }     // keep first index on ties
        }
    }

    // ---- argmin across the 16 lanes of each half (lexicographic: area, index) ----
#pragma unroll
    for (int off = 1; off < 16; off <<= 1) {
#pragma unroll
        for (int v = 0; v < 8; ++v) {
            float ao = __shfl_xor(bA[v], off, 32);
            int   io = __shfl_xor(bI[v], off, 32);
            if (ao < bA[v] || (ao == bA[v] && io < bI[v])) { bA[v] = ao; bI[v] = io; }
        }
    }

    // ---- writers: lanes 0-7 emit rows 0-7, lanes 16-23 emit rows 8-15 ----
    const int  slot   = lane & 7;
    const bool writer = (l16 < 8);
    float wA = bA[0];
    int   wI = bI[0];
#pragma unroll
    for (int v = 1; v < 8; ++v)
        if (slot == v) { wA = bA[v]; wI = bI[v]; }

    const int mOut = slot + 8 * half;
    const int loc  = tIn * 16 + mOut;
    if (writer && loc < hw) {
        const bool   posAny = (wA < INF_AREA);
        const float4 bx     = sBox[wI];
        const int    cls    = sCls[wI];
        const float  x   = (float)(loc % w) * stride + stride * 0.5f;
        const float  y   = (float)(loc / w) * stride + stride * 0.5f;
        const float  inv = 1.0f / stride;                 // stride is a power of 2: exact
        float l  = (x - bx.x) * inv;
        float tt = (y - bx.y) * inv;
        float r  = (bx.z - x) * inv;
        float bb = (bx.w - y) * inv;
        float lrmin = fminf(l, r),  lrmax = fmaxf(l, r);
        float tbmin = fminf(tt, bb), tbmax = fmaxf(tt, bb);
        float ctr = sqrtf(fmaxf(lrmin * tbmin, 0.0f) / fmaxf(lrmax * tbmax, 1e-10f));

        float co = (float)cls, r0 = l, r1 = tt, r2 = r, r3 = bb, ct = ctr;
        if (!posAny) { co = 0.0f; r0 = r1 = r2 = r3 = -1.0f; ct = -1.0f; }

        const int bl = b * hw + loc;
        out[kClsOff[level] + bl] = co;
        float* rp = out + kRegOff[level] + (size_t)bl * 4;
        rp[0] = r0; rp[1] = r1; rp[2] = r2; rp[3] = r3;
        out[kCtrOff[level] + bl] = ct;
    }
}

extern "C" void kernel_launch(void* const* d_in, const int* in_sizes, int n_in,
                              void* d_out, int out_size, void* d_ws, size_t ws_size,
                              hipStream_t stream) {
    // input order: feat0..feat4 (unused), cls_ids, boxes
    const int*   cls_ids = (const int*)d_in[5];
    const float* boxes   = (const float*)d_in[6];
    float*       out     = (float*)d_out;

    dim3 grid(8 * TILES_PER_BATCH);  // 8536 wave-sized blocks
    dim3 block(32);
    hipLaunchKernelGGL(fcos_target_wmma, grid, block, 0, stream, boxes, cls_ids, out);
}